// MegaLocMPS_14276471292147
// MI455X (gfx1250) — compile-verified
//
#include <hip/hip_runtime.h>

typedef float v2f __attribute__((ext_vector_type(2)));
typedef float v8f __attribute__((ext_vector_type(8)));

#define THRESH 0.7f

// Full-precision f32 WMMA: D(16x16) = A(16x4) * B(4x16) + C
// A frag: lanes 0-15 -> row M=lane, K=k0,k0+1 ; lanes 16-31 -> K=k0+2,k0+3
// B frag (row-striped): lanes 0-15 -> col N=lane, rows k0 (x) / k0+1 (y);
//                       lanes 16-31 -> rows k0+2 / k0+3
__device__ __forceinline__ v8f wmma_f32(v2f a, v2f b, v8f c) {
    return __builtin_amdgcn_wmma_f32_16x16x4_f32(
        /*neg_a=*/false, a, /*neg_b=*/false, b,
        /*c_mod=*/(short)0, c, /*reuse_a=*/false, /*reuse_b=*/false);
}

// C[M,N] = act(A[M,K] @ B[K,N] + bias), one wave per 16x16 tile.
__global__ void gemm_bias_kernel(const float* __restrict__ A,
                                 const float* __restrict__ B,
                                 const float* __restrict__ bias,
                                 float* __restrict__ C,
                                 int M, int N, int K, int relu) {
    int gtid = blockIdx.x * blockDim.x + threadIdx.x;
    int wave = gtid >> 5;
    int lane = threadIdx.x & 31;
    int ntiles = N >> 4;
    int mt = wave / ntiles;
    int nt = wave - mt * ntiles;
    if (mt * 16 >= M) return;                 // wave-uniform guard (EXEC stays all-1)
    int m0 = mt * 16, n0 = nt * 16;
    int hh = lane >> 4;                       // half-wave select
    int l  = lane & 15;

    const float* arow = A + (size_t)(m0 + l) * K;
    const float* bcol = B + (n0 + l);

    v8f c = {};
    for (int k0 = 0; k0 < K; k0 += 4) {
        int ka = k0 + 2 * hh;
        v2f a = *(const v2f*)(arow + ka);     // 8B-aligned: ka is even
        v2f b;
        b.x = bcol[(size_t)ka * N];
        b.y = bcol[(size_t)(ka + 1) * N];
        c = wmma_f32(a, b, c);
    }
#pragma unroll
    for (int r = 0; r < 8; ++r) {
        int row = m0 + r + 8 * hh;
        int col = n0 + l;
        float v = c[r] + bias[col];
        if (relu) v = fmaxf(v, 0.0f);
        C[(size_t)row * N + col] = v;
    }
}

// One block per batch row: norm over concat(ones[16384], g[256]); write feats row.
__global__ void feats_kernel(const float* __restrict__ g,
                             float* __restrict__ feats,
                             float* __restrict__ inv_out) {
    __shared__ float red[256];
    __shared__ float invs;
    int b = blockIdx.x, t = threadIdx.x;
    float v = g[b * 256 + t];
    red[t] = v * v;
    __syncthreads();
    for (int s = 128; s > 0; s >>= 1) {
        if (t < s) red[t] += red[t + s];
        __syncthreads();
    }
    if (t == 0) {
        float nrm = fmaxf(sqrtf(16384.0f + red[0]), 1e-12f);
        invs = 1.0f / nrm;
        inv_out[b] = invs;
    }
    __syncthreads();
    float inv = invs;
    float* row = feats + (size_t)b * 16640;
    for (int i = t; i < 16384; i += 256) row[i] = inv;   // ones-block, normalized
    row[16384 + t] = v * inv;                            // global-token block
}

// sim = feats@feats^T via g@g^T (+16384 from the ones block), then
// threshold + top-k symmetric scatter + diagonal, all in one workgroup.
__global__ void sim_mask_kernel(const float* __restrict__ g,
                                const float* __restrict__ inv,
                                const int* __restrict__ kptr,
                                float* __restrict__ mask) {
    __shared__ float simL[64 * 64];
    __shared__ float maskL[64 * 64];
    int tid  = threadIdx.x;
    int wave = tid >> 5;
    int lane = tid & 31;
    int hh = lane >> 4;
    int l  = lane & 15;
    int m0 = (wave >> 2) * 16;
    int n0 = (wave & 3) * 16;

    const float* arow = g + (size_t)(m0 + l) * 256;
    const float* brow = g + (size_t)(n0 + l) * 256;   // B = g^T -> rows of g

    v8f c = {};
    for (int k0 = 0; k0 < 256; k0 += 4) {
        v2f a = *(const v2f*)(arow + k0 + 2 * hh);
        v2f b = *(const v2f*)(brow + k0 + 2 * hh);
        c = wmma_f32(a, b, c);
    }
#pragma unroll
    for (int r = 0; r < 8; ++r) {
        int i = m0 + r + 8 * hh;
        int j = n0 + l;
        float s = (c[r] + 16384.0f) * inv[i] * inv[j];
        simL[i * 64 + j]  = s;
        maskL[i * 64 + j] = (s > THRESH) ? 1.0f : 0.0f;
    }
    __syncthreads();

    if (tid < 64) {
        int i = tid;
        float* row = simL + i * 64;
        int k = kptr[0];
        k = k < 0 ? 0 : (k > 64 ? 64 : k);
        for (int t = 0; t < k; ++t) {
            float best = -3.402823466e38f;
            int bi = 0;
            for (int j = 0; j < 64; ++j) {            // strict > = lowest-index tie-break
                float v = row[j];
                if (v > best) { best = v; bi = j; }
            }
            row[bi] = -3.402823466e38f;
            maskL[i * 64 + bi] = 1.0f;
            maskL[bi * 64 + i] = 1.0f;                // benign 1.0-write races
        }
        maskL[i * 64 + i] = 1.0f;
    }
    __syncthreads();

    for (int idx = tid; idx < 4096; idx += 512) mask[idx] = maskL[idx];
}

extern "C" void kernel_launch(void* const* d_in, const int* in_sizes, int n_in,
                              void* d_out, int out_size, void* d_ws, size_t ws_size,
                              hipStream_t stream) {
    (void)in_sizes; (void)n_in; (void)out_size; (void)ws_size;
    // Input order: 0 local_features(unused), 1 global_token, 2..5 local MLP (unused:
    // softmax(axis=1).sum(axis=1) == 1 identically), 6 W1_global, 7 b1_global,
    // 8 W2_global, 9 b2_global, 10 k_nearest.
    const float* global_token = (const float*)d_in[1];
    const float* W1g = (const float*)d_in[6];
    const float* b1g = (const float*)d_in[7];
    const float* W2g = (const float*)d_in[8];
    const float* b2g = (const float*)d_in[9];
    const int*   kp  = (const int*)d_in[10];

    float* feats = (float*)d_out;                         // 64 x 16640
    float* mask  = feats + (size_t)64 * 16640;            // 64 x 64

    float* h    = (float*)d_ws;                           // 64 x 512
    float* g    = h + 64 * 512;                           // 64 x 256
    float* invs = g + 64 * 256;                           // 64

    // h = relu(global_token @ W1_global + b1): M=64,N=512,K=768 -> 128 waves
    gemm_bias_kernel<<<16, 256, 0, stream>>>(global_token, W1g, b1g, h, 64, 512, 768, 1);
    // g = h @ W2_global + b2: M=64,N=256,K=512 -> 64 waves
    gemm_bias_kernel<<<8, 256, 0, stream>>>(h, W2g, b2g, g, 64, 256, 512, 0);
    // feats rows + per-row inverse norms
    feats_kernel<<<64, 256, 0, stream>>>(g, feats, invs);
    // covisibility mask (sim + threshold + top-k + diag) in one workgroup
    sim_mask_kernel<<<1, 512, 0, stream>>>(g, invs, kp, mask);
}